// GCN_22153441313372
// MI455X (gfx1250) — compile-verified
//
#include <hip/hip_runtime.h>
#include <hip/hip_bf16.h>

// ---------------------------------------------------------------------------
// GCN 2-layer message passing for MI455X (gfx1250, wave32).
// GEMMs run on V_WMMA_F32_16X16X4_F32 (exact f32 path; workload is
// HBM/atomic-bound so f32 matrix ops already exceed the bandwidth roofline).
// Tile staging uses GLOBAL_LOAD_ASYNC_TO_LDS_B128 (ASYNCcnt data-mover path).
// ---------------------------------------------------------------------------

typedef float v2f __attribute__((ext_vector_type(2)));
typedef float v8f __attribute__((ext_vector_type(8)));
typedef int   v4i __attribute__((vector_size(4 * sizeof(int))));

#define N_NODES 40000
#define N_EDGES 640000
#define NODE_IN 128
#define EDGE_IN 64
#define HID     128

// --- async global->LDS helpers (gfx1250 data-mover path, guarded) ----------
__device__ __forceinline__ void stage_b128(const float* g, float* l) {
#if defined(__gfx1250__) && __has_builtin(__builtin_amdgcn_global_load_async_to_lds_b128)
  // Param types per clang diagnostic: (AS1 v4i*, AS3 v4i*, imm offset, imm cpol)
  __builtin_amdgcn_global_load_async_to_lds_b128(
      (__attribute__((address_space(1))) v4i*)g,
      (__attribute__((address_space(3))) v4i*)l,
      /*imm offset=*/0, /*cpol=*/0);
#else
  *(float4*)l = *(const float4*)g;
#endif
}

__device__ __forceinline__ void stage_wait() {
#if defined(__gfx1250__) && __has_builtin(__builtin_amdgcn_global_load_async_to_lds_b128)
#if __has_builtin(__builtin_amdgcn_s_wait_asynccnt)
  __builtin_amdgcn_s_wait_asynccnt(0);
#else
  asm volatile("s_wait_asynccnt 0x0" ::: "memory");
#endif
#endif
}

// ---------------------------------------------------------------------------
// H[row0..row0+16) x 128  =  X[16 x 128] @ W[128 x 128] + bias
// 8 waves per block, wave w owns columns [16w, 16w+16). K = 128.
// ---------------------------------------------------------------------------
__global__ __launch_bounds__(256) void node_gemm_wmma(
    const float* __restrict__ X,     // [N, 128]
    const float* __restrict__ W,     // [128, 128]
    const float* __restrict__ bias,  // [128]
    float* __restrict__ H)           // [N, 128]
{
  constexpr int K  = 128;
  constexpr int KP = K + 4;              // padded LDS row stride (dwords): bank = 4m+k
  __shared__ float ldsX[16 * KP];

  const int tid  = threadIdx.x;
  const int row0 = blockIdx.x * 16;

  // Stage the 16x128 X tile via the async data mover (b128 per lane).
  for (int i = tid * 4; i < 16 * K; i += 256 * 4) {
    const int r = i / K, c = i % K;
    stage_b128(X + (size_t)(row0 + r) * K + c, &ldsX[r * KP + c]);
  }
  stage_wait();
  __syncthreads();

  const int lane = tid & 31;
  const int wave = tid >> 5;
  const int col0 = wave * 16;
  const int hl   = lane & 15;
  const bool hi  = lane >= 16;

  v8f acc = {};
#pragma unroll
  for (int k0 = 0; k0 < K; k0 += 4) {
    const int ka = k0 + (hi ? 2 : 0);
    v2f a, b;
    // A 16x4 layout: lanes 0-15 hold {K=k0,k0+1} of row hl; lanes 16-31 hold {k0+2,k0+3}
    a.x = ldsX[hl * KP + ka];
    a.y = ldsX[hl * KP + ka + 1];
    // B 4x16 layout mirrors the K split across lane halves.
    const float* wp = W + (size_t)ka * HID + col0 + hl;
    b.x = wp[0];
    b.y = wp[HID];
    acc = __builtin_amdgcn_wmma_f32_16x16x4_f32(
        false, a, false, b, (short)0, acc, false, false);
  }

  const float bv = bias[col0 + hl];
  float* out = H + (size_t)(row0 + (hi ? 8 : 0)) * HID + col0 + hl;
#pragma unroll
  for (int v = 0; v < 8; ++v)           // C/D: VGPR v -> row v (lo lanes) / v+8 (hi lanes)
    out[(size_t)v * HID] = acc[v] + bv;
}

// ---------------------------------------------------------------------------
// Fused edge MLP + message + scatter-add for a 16-edge tile:
//   e   = EF[16 x 64] @ We[64 x 128] + be          (WMMA)
//   msg = e + Hnode[src]                           (gather)
//   Agg[dst] += msg                                (global_atomic_add_f32)
// ---------------------------------------------------------------------------
__global__ __launch_bounds__(256) void edge_msg_wmma(
    const float* __restrict__ EF,     // [E, 64]
    const float* __restrict__ We,     // [64, 128]
    const float* __restrict__ be,     // [128]
    const float* __restrict__ Hnode,  // [N, 128]
    const int* __restrict__ src,      // [E]
    const int* __restrict__ dst,      // [E]
    float* __restrict__ Agg)          // [N, 128], pre-zeroed
{
  constexpr int K  = 64;
  constexpr int KP = K + 4;
  __shared__ float ldsE[16 * KP];
  __shared__ int   ldsSrc[16];
  __shared__ int   ldsDst[16];

  const int tid = threadIdx.x;
  const int e0  = blockIdx.x * 16;

  // 16x64 tile = exactly one b128 per thread.
  {
    const int r = tid / (K / 4), c4 = tid % (K / 4);
    stage_b128(EF + (size_t)(e0 + r) * K + c4 * 4, &ldsE[r * KP + c4 * 4]);
  }
  if (tid < 16) {
    ldsSrc[tid] = src[e0 + tid];
    ldsDst[tid] = dst[e0 + tid];
  }
  stage_wait();
  __syncthreads();

  const int lane = tid & 31;
  const int wave = tid >> 5;
  const int col0 = wave * 16;
  const int hl   = lane & 15;
  const bool hi  = lane >= 16;

  v8f acc = {};
#pragma unroll
  for (int k0 = 0; k0 < K; k0 += 4) {
    const int ka = k0 + (hi ? 2 : 0);
    v2f a, b;
    a.x = ldsE[hl * KP + ka];
    a.y = ldsE[hl * KP + ka + 1];
    const float* wp = We + (size_t)ka * HID + col0 + hl;
    b.x = wp[0];
    b.y = wp[HID];
    acc = __builtin_amdgcn_wmma_f32_16x16x4_f32(
        false, a, false, b, (short)0, acc, false, false);
  }

  const float bv  = be[col0 + hl];
  const int   col = col0 + hl;
#pragma unroll
  for (int v = 0; v < 8; ++v) {
    const int r = v + (hi ? 8 : 0);          // local edge row
    const int s = ldsSrc[r];
    const int d = ldsDst[r];
    const float val = acc[v] + bv + Hnode[(size_t)s * HID + col];
    atomicAdd(&Agg[(size_t)d * HID + col], val);
  }
}

// ---------------------------------------------------------------------------
__global__ __launch_bounds__(256) void relu_kernel(float* __restrict__ x, int n4)
{
  const int i = blockIdx.x * blockDim.x + threadIdx.x;
  if (i < n4) {
    float4 v = ((float4*)x)[i];
    v.x = fmaxf(v.x, 0.f); v.y = fmaxf(v.y, 0.f);
    v.z = fmaxf(v.z, 0.f); v.w = fmaxf(v.w, 0.f);
    ((float4*)x)[i] = v;
  }
}

// ---------------------------------------------------------------------------
// Global add-pool over rows whose features are not all identical.
// One wave32 per row; lane covers columns {lane, lane+32, lane+64, lane+96}.
// ---------------------------------------------------------------------------
__global__ __launch_bounds__(256) void pool_kernel(
    const float* __restrict__ X, float* __restrict__ out, int n_rows)
{
  __shared__ float sacc[128];
  const int tid = threadIdx.x;
  if (tid < 128) sacc[tid] = 0.f;
  __syncthreads();

  const int lane = tid & 31;
  const int wave = tid >> 5;
  float s0 = 0.f, s1 = 0.f, s2 = 0.f, s3 = 0.f;

  for (int row = blockIdx.x * 8 + wave; row < n_rows; row += gridDim.x * 8) {
    const float* rp = X + (size_t)row * 128;
    const float a = rp[lane], b = rp[lane + 32], c = rp[lane + 64], d = rp[lane + 96];
    float mx = fmaxf(fmaxf(a, b), fmaxf(c, d));
    float mn = fminf(fminf(a, b), fminf(c, d));
#pragma unroll
    for (int off = 16; off > 0; off >>= 1) {   // wave32 reduction
      mx = fmaxf(mx, __shfl_xor(mx, off, 32));
      mn = fminf(mn, __shfl_xor(mn, off, 32));
    }
    if (mx != mn) { s0 += a; s1 += b; s2 += c; s3 += d; }
  }
  atomicAdd(&sacc[lane],      s0);
  atomicAdd(&sacc[lane + 32], s1);
  atomicAdd(&sacc[lane + 64], s2);
  atomicAdd(&sacc[lane + 96], s3);
  __syncthreads();
  if (tid < 128) atomicAdd(&out[tid], sacc[tid]);
}

// ---------------------------------------------------------------------------
extern "C" void kernel_launch(void* const* d_in, const int* in_sizes, int n_in,
                              void* d_out, int out_size, void* d_ws, size_t ws_size,
                              hipStream_t stream)
{
  const float* node_feature = (const float*)d_in[0];
  const int*   edge_index   = (const int*)d_in[1];   // [2, E]
  const float* edge_feature = (const float*)d_in[2];
  const float* W1n = (const float*)d_in[3];
  const float* b1n = (const float*)d_in[4];
  const float* W1e = (const float*)d_in[5];
  const float* b1e = (const float*)d_in[6];
  const float* W2n = (const float*)d_in[7];
  const float* b2n = (const float*)d_in[8];
  const float* W2e = (const float*)d_in[9];
  const float* b2e = (const float*)d_in[10];

  const int* src = edge_index;             // row 0: source nodes
  const int* dst = edge_index + N_EDGES;   // row 1: target nodes

  float* out  = (float*)d_out;
  const size_t NH = (size_t)N_NODES * HID;
  float* h    = (float*)d_ws;   // [N,128] hidden (reused by both layers)
  float* agg1 = h + NH;         // [N,128] layer-1 aggregation (then relu'd)
  float* agg2 = agg1 + NH;      // [N,128] layer-2 aggregation

  // Zero accumulation targets (agg1+agg2 are contiguous) and the output.
  (void)hipMemsetAsync(agg1, 0, 2 * NH * sizeof(float), stream);
  (void)hipMemsetAsync(d_out, 0, (size_t)HID * sizeof(float), stream);

  const int node_blocks = N_NODES / 16;   // 2500
  const int edge_blocks = N_EDGES / 16;   // 40000

  // Layer 1
  node_gemm_wmma<<<node_blocks, 256, 0, stream>>>(node_feature, W1n, b1n, h);
  edge_msg_wmma<<<edge_blocks, 256, 0, stream>>>(edge_feature, W1e, b1e, h,
                                                 src, dst, agg1);
  const int n4 = (int)(NH / 4);
  relu_kernel<<<(n4 + 255) / 256, 256, 0, stream>>>(agg1, n4);

  // Layer 2
  node_gemm_wmma<<<node_blocks, 256, 0, stream>>>(agg1, W2n, b2n, h);
  edge_msg_wmma<<<edge_blocks, 256, 0, stream>>>(edge_feature, W2e, b2e, h,
                                                 src, dst, agg2);

  // Global pool with non-constant-row mask.
  pool_kernel<<<512, 256, 0, stream>>>(agg2, out, N_NODES);
}